// ComnetModel_38723425141291
// MI455X (gfx1250) — compile-verified
//
#include <hip/hip_runtime.h>
#include <hip/hip_bf16.h>

// ---------------------------------------------------------------------------
// RouteNet-style ComnetModel for MI455X (gfx1250, wave32).
//
// Design rationale (compile-only; reasoned from MI455X specs):
//  * Total dense FLOPs ~0.8 GF -> compute is free. Bottleneck is 3.4M random
//    16B gathers + ~7M f32 scatter-atomics per T-step, all hitting the 192MB
//    L2 (state tables are <3MB). So: scalar one-thread-per-path GRU with
//    LDS-staged weights, float4 state loads, global f32 atomics for the
//    segment sums. 8 kernel launches give the required grid-wide ordering.
//  * The readout MLP (2->8->8->1) is the only cleanly dense stage: mapped to
//    V_WMMA_F32_16X16X4_F32 (exact fp32) in transposed orientation
//    D = W^T x X^T so each lane carries all 8 features of one path in the
//    v8f accumulator; SELU + the chained 8x8 layer stay in-lane. All padding
//    masks are branchless (v_cndmask) so EXEC stays all-ones around WMMA.
// ---------------------------------------------------------------------------

typedef __attribute__((ext_vector_type(2))) float v2f;
typedef __attribute__((ext_vector_type(8))) float v8f;

#define LINK_DIM 4
#define PATH_DIM 2
#define K_LINKS 8
#define K_NODES 9

__device__ __forceinline__ float sigmoidf_(float x) {
    return 1.0f / (1.0f + __expf(-x));
}
__device__ __forceinline__ float tanhf_(float x) {
    return 1.0f - 2.0f / (__expf(2.0f * x) + 1.0f);
}
// Branchless SELU: for x>0 the exp term is exp(0)-1 == 0; for x<=0 the relu
// term is 0. Identical result, no exec-mask divergence.
__device__ __forceinline__ float seluf_(float x) {
    const float scale = 1.0507009873554805f;
    const float scale_alpha = 1.7580993408473766f;  // scale*alpha
    return scale * fmaxf(x, 0.0f) + scale_alpha * (__expf(fminf(x, 0.0f)) - 1.0f);
}

// ---------------------------------------------------------------------------
// Init: build states from inputs, zero message accumulators.
// link_state[n_links][4], node_state[n_nodes][4], path_state[n_paths][2]
// ---------------------------------------------------------------------------
__global__ __launch_bounds__(256) void comnet_init_kernel(
    const float* __restrict__ cap, const float* __restrict__ qs,
    const float* __restrict__ traffic,
    float* __restrict__ link_state, float* __restrict__ node_state,
    float* __restrict__ path_state,
    float* __restrict__ m_links, float* __restrict__ m_nodes,
    int n_links, int n_nodes, int n_paths)
{
    int i = blockIdx.x * blockDim.x + threadIdx.x;
    if (i < n_links) {
        *(float4*)(link_state + 4 * i) = make_float4(cap[i], 0.f, 0.f, 0.f);
        m_links[2 * i] = 0.f;
        m_links[2 * i + 1] = 0.f;
    }
    if (i < n_nodes) {
        *(float4*)(node_state + 4 * i) = make_float4(qs[i], 0.f, 0.f, 0.f);
        m_nodes[2 * i] = 0.f;
        m_nodes[2 * i + 1] = 0.f;
    }
    if (i < n_paths) {
        path_state[2 * i] = traffic[i];
        path_state[2 * i + 1] = 0.f;
    }
}

// One path-GRU step (input dim 4, hidden dim 2). Weights in LDS.
__device__ __forceinline__ void path_gru_step(
    float4 x, float& h0, float& h1,
    const float* __restrict__ sK,   // (4,6) row-major
    const float* __restrict__ sR,   // (2,6) row-major
    const float* __restrict__ sB)   // (6,)
{
    float xk[6];
#pragma unroll
    for (int j = 0; j < 6; ++j)
        xk[j] = sB[j] + x.x * sK[j] + x.y * sK[6 + j] +
                x.z * sK[12 + j] + x.w * sK[18 + j];

    float z0 = sigmoidf_(xk[0] + h0 * sR[0] + h1 * sR[6 + 0]);
    float z1 = sigmoidf_(xk[1] + h0 * sR[1] + h1 * sR[6 + 1]);
    float r0 = sigmoidf_(xk[2] + h0 * sR[2] + h1 * sR[6 + 2]);
    float r1 = sigmoidf_(xk[3] + h0 * sR[3] + h1 * sR[6 + 3]);
    float hh0 = tanhf_(xk[4] + r0 * h0 * sR[4] + r1 * h1 * sR[6 + 4]);
    float hh1 = tanhf_(xk[5] + r0 * h0 * sR[5] + r1 * h1 * sR[6 + 5]);
    h0 = z0 * h0 + (1.0f - z0) * hh0;
    h1 = z1 * h1 + (1.0f - z1) * hh1;
}

// ---------------------------------------------------------------------------
// Part 1: per-path GRU over the interleaved node/link sequence
// (node, link, node, ..., node: 9 nodes + 8 links = 17 steps, always fully
// valid). Each step's new hidden state is the RNN output -> scatter-add into
// the link/node message accumulators immediately (segment_sum via atomics).
// The alternation is expressed structurally: 8 x (node step, link step) + 1
// final node step -> no parity branches inside the loop.
// ---------------------------------------------------------------------------
__global__ __launch_bounds__(256) void comnet_path_rnn_kernel(
    const float* __restrict__ node_state, const float* __restrict__ link_state,
    float* __restrict__ path_state,
    const int* __restrict__ nodes, const int* __restrict__ links,
    float* __restrict__ m_nodes, float* __restrict__ m_links,
    const float* __restrict__ pk, const float* __restrict__ pr,
    const float* __restrict__ pb, int n_paths)
{
    __shared__ float sK[24], sR[12], sB[6];
    int t = threadIdx.x;
    if (t < 24) sK[t] = pk[t];
    if (t < 12) sR[t] = pr[t];
    if (t < 6)  sB[t] = pb[t];
    __syncthreads();

    int p = blockIdx.x * blockDim.x + t;
    if (p >= n_paths) return;

    float h0 = path_state[2 * p];
    float h1 = path_state[2 * p + 1];
    const int* pn = nodes + p * K_NODES;
    const int* pl = links + p * K_LINKS;

    for (int s = 0; s < K_LINKS; ++s) {
        // node hop (even position 2s)
        int ni = pn[s];
        float4 xn = *(const float4*)(node_state + 4 * ni);
        path_gru_step(xn, h0, h1, sK, sR, sB);
        atomicAdd(m_nodes + 2 * ni, h0);
        atomicAdd(m_nodes + 2 * ni + 1, h1);
        // link hop (odd position 2s+1)
        int li = pl[s];
        float4 xl = *(const float4*)(link_state + 4 * li);
        path_gru_step(xl, h0, h1, sK, sR, sB);
        atomicAdd(m_links + 2 * li, h0);
        atomicAdd(m_links + 2 * li + 1, h1);
    }
    // final node hop (position 16)
    int ni = pn[K_LINKS];
    float4 xn = *(const float4*)(node_state + 4 * ni);
    path_gru_step(xn, h0, h1, sK, sR, sB);
    atomicAdd(m_nodes + 2 * ni, h0);
    atomicAdd(m_nodes + 2 * ni + 1, h1);

    path_state[2 * p] = h0;
    path_state[2 * p + 1] = h1;
}

// ---------------------------------------------------------------------------
// Part 2: link GRU + node GRU updates (input dim 2, hidden dim 4), fused into
// one kernel; also re-zeroes the message accumulators for the next T-step.
// ---------------------------------------------------------------------------
__global__ __launch_bounds__(256) void comnet_edge_node_update_kernel(
    float* __restrict__ link_state, float* __restrict__ node_state,
    float* __restrict__ m_links, float* __restrict__ m_nodes,
    const float* __restrict__ ek, const float* __restrict__ er,
    const float* __restrict__ ebias,
    const float* __restrict__ nk, const float* __restrict__ nr,
    const float* __restrict__ nbias,
    int n_links, int n_nodes)
{
    // shared layout: [0..23] ek (2x12), [24..71] er (4x12), [72..83] eb,
    //                [84..107] nk,      [108..155] nr,      [156..167] nb
    __shared__ float s[168];
    int t = threadIdx.x;
    if (t < 24)       s[t] = ek[t];
    else if (t < 72)  s[t] = er[t - 24];
    else if (t < 84)  s[t] = ebias[t - 72];
    else if (t < 108) s[t] = nk[t - 84];
    else if (t < 156) s[t] = nr[t - 108];
    else if (t < 168) s[t] = nbias[t - 156];
    __syncthreads();

    int i = blockIdx.x * blockDim.x + t;
    bool isLink = i < n_links;
    int idx = isLink ? i : i - n_links;
    if (!isLink && idx >= n_nodes) return;

    const float* K = isLink ? s : s + 84;   // (2,12)
    const float* R = K + 24;                // (4,12)
    const float* B = K + 72;                // (12,)
    float* st = (isLink ? link_state : node_state) + 4 * idx;
    float* m  = (isLink ? m_links   : m_nodes)   + 2 * idx;

    float m0 = m[0], m1 = m[1];
    m[0] = 0.f; m[1] = 0.f;  // reset accumulator for next T-iteration

    float4 hv = *(const float4*)st;
    float h[4] = {hv.x, hv.y, hv.z, hv.w};

    float xk[12];
#pragma unroll
    for (int j = 0; j < 12; ++j)
        xk[j] = B[j] + m0 * K[j] + m1 * K[12 + j];

    float z[4], r[4];
#pragma unroll
    for (int j = 0; j < 4; ++j) {
        float a = xk[j], b = xk[4 + j];
#pragma unroll
        for (int k = 0; k < 4; ++k) {
            a += h[k] * R[k * 12 + j];
            b += h[k] * R[k * 12 + 4 + j];
        }
        z[j] = sigmoidf_(a);
        r[j] = sigmoidf_(b);
    }
    float rh[4];
#pragma unroll
    for (int k = 0; k < 4; ++k) rh[k] = r[k] * h[k];
#pragma unroll
    for (int j = 0; j < 4; ++j) {
        float a = xk[8 + j];
#pragma unroll
        for (int k = 0; k < 4; ++k) a += rh[k] * R[k * 12 + 8 + j];
        float hh = tanhf_(a);
        h[j] = z[j] * h[j] + (1.0f - z[j]) * hh;
    }
    *(float4*)st = make_float4(h[0], h[1], h[2], h[3]);
}

// ---------------------------------------------------------------------------
// Readout MLP via V_WMMA_F32_16X16X4_F32 (exact fp32).
// Orientation: D = W^T (16x4 padded A) x X^T (4x16 padded B).
// ISA 7.12.2 layouts assumed:
//   A 16x4 : lanes 0-15 hold M=lane with {v0=K0, v1=K1}; lanes 16-31 {K2,K3}
//   B 4x16 : lanes 0-15 hold N=lane with {v0=K0, v1=K1}; lanes 16-31 {K2,K3}
//   D 16x16: vgpr r, lanes 0-15 -> row r, lanes 16-31 -> row 8+r (col = lane&15)
// => after D = W^T x X^T, lane n (n<16) holds ALL 8 features of path n in its
//    v8f accumulator; rows 8..15 (hi lanes) are zero padding.
// 16 paths per wave; guard is wave-uniform and all padding masks are value
// selects (v_cndmask), so EXEC is all-ones at every WMMA as required.
// ---------------------------------------------------------------------------
__global__ __launch_bounds__(256) void comnet_readout_wmma_kernel(
    const float* __restrict__ path_state,
    const float* __restrict__ w1, const float* __restrict__ b1,  // (2,8),(8)
    const float* __restrict__ w2, const float* __restrict__ b2,  // (8,8),(8)
    const float* __restrict__ w3, const float* __restrict__ b3,  // (8,1),(1)
    float* __restrict__ out, int n_paths)
{
    int tid = blockIdx.x * blockDim.x + threadIdx.x;
    int wave = tid >> 5;
    int lane = threadIdx.x & 31;
    int base = wave * 16;
    if (base >= n_paths) return;  // uniform across the wave

    int l = lane & 15;
    bool hi = lane >= 16;
    bool in8 = (l < 8);
    int l8 = l & 7;               // clamped index: loads always valid
    int p = base + l;
    int pc = p < n_paths ? p : n_paths - 1;

    // ---- Unconditional weight/bias/state loads (broadcast, L2-hot) ----
    float w1k0 = w1[l8], w1k1 = w1[8 + l8];
    float w2r0 = w2[(hi ? 2 : 0) * 8 + l8];
    float w2r1 = w2[(hi ? 3 : 1) * 8 + l8];
    float w2r2 = w2[(hi ? 6 : 4) * 8 + l8];
    float w2r3 = w2[(hi ? 7 : 5) * 8 + l8];
    float ps0 = path_state[2 * pc];
    float ps1 = path_state[2 * pc + 1];
    float b1v[8], b2v[8], w3v[8];
#pragma unroll
    for (int q = 0; q < 8; ++q) { b1v[q] = b1[q]; b2v[q] = b2[q]; w3v[q] = w3[q]; }
    float b3v = b3[0];

    // ---- Layer 1: H1^T = W1^T x PS^T + b1 ----
    // A = W1^T padded to 16x4: A[m][k] = w1[k*8+m] (m<8, k<2)
    v2f a1;
    a1.x = (!hi && in8) ? w1k0 : 0.0f;  // K=0
    a1.y = (!hi && in8) ? w1k1 : 0.0f;  // K=1
    // B = PS^T padded to 4x16: B[k][n] = ps[n][k] (k<2)
    v2f bb;
    bb.x = hi ? 0.0f : ps0;
    bb.y = hi ? 0.0f : ps1;
    v8f c;
#pragma unroll
    for (int q = 0; q < 8; ++q) c[q] = hi ? 0.0f : b1v[q];
    c = __builtin_amdgcn_wmma_f32_16x16x4_f32(false, a1, false, bb,
                                              (short)0, c, false, false);
#pragma unroll
    for (int q = 0; q < 8; ++q) c[q] = seluf_(c[q]);  // selu(0)==0: padding safe

    // ---- Layer 2: H2^T = W2^T x H1^T + b2, K=8 via two K=4 WMMAs ----
    v2f a2lo, a2hi;
    a2lo.x = in8 ? w2r0 : 0.0f;
    a2lo.y = in8 ? w2r1 : 0.0f;
    a2hi.x = in8 ? w2r2 : 0.0f;
    a2hi.y = in8 ? w2r3 : 0.0f;
    // B operands: hi lanes need features 2,3 (resp. 6,7) of path (lane-16),
    // which live in lane (lane-16)'s accumulator -> cross-half shuffles.
    float s2 = __shfl(c[2], l);
    float s3 = __shfl(c[3], l);
    float s6 = __shfl(c[6], l);
    float s7 = __shfl(c[7], l);
    v2f b2lo, b2hi;
    b2lo.x = hi ? s2 : c[0];
    b2lo.y = hi ? s3 : c[1];
    b2hi.x = hi ? s6 : c[4];
    b2hi.y = hi ? s7 : c[5];
    v8f d;
#pragma unroll
    for (int q = 0; q < 8; ++q) d[q] = hi ? 0.0f : b2v[q];
    d = __builtin_amdgcn_wmma_f32_16x16x4_f32(false, a2lo, false, b2lo,
                                              (short)0, d, false, false);
    d = __builtin_amdgcn_wmma_f32_16x16x4_f32(false, a2hi, false, b2hi,
                                              (short)0, d, false, false);
#pragma unroll
    for (int q = 0; q < 8; ++q) d[q] = seluf_(d[q]);

    // ---- Layer 3: 8 -> 1 dot product, in-lane ----
    float o = b3v;
#pragma unroll
    for (int q = 0; q < 8; ++q) o += d[q] * w3v[q];
    if (!hi && p < n_paths) out[p] = o;
}

// ---------------------------------------------------------------------------
extern "C" void kernel_launch(void* const* d_in, const int* in_sizes, int n_in,
                              void* d_out, int out_size, void* d_ws, size_t ws_size,
                              hipStream_t stream) {
    (void)n_in; (void)out_size; (void)ws_size;

    const float* cap     = (const float*)d_in[0];   // link_capacity [n_links]
    const float* qs      = (const float*)d_in[1];   // queue_sizes   [n_nodes]
    const float* traffic = (const float*)d_in[2];   // traffic       [n_paths]
    const int*   links   = (const int*)d_in[3];     // [n_paths*8]
    // d_in[4] link_paths, d_in[5] link_seqs: implied by layout, unused
    const int*   nodes   = (const int*)d_in[6];     // [n_paths*9]
    // d_in[7] node_paths, d_in[8] node_seqs: unused
    const float* pk = (const float*)d_in[9];
    const float* pr = (const float*)d_in[10];
    const float* pb = (const float*)d_in[11];
    const float* ek = (const float*)d_in[12];
    const float* er = (const float*)d_in[13];
    const float* eb = (const float*)d_in[14];
    const float* nk = (const float*)d_in[15];
    const float* nr = (const float*)d_in[16];
    const float* nb = (const float*)d_in[17];
    const float* w1 = (const float*)d_in[18];
    const float* b1 = (const float*)d_in[19];
    const float* w2 = (const float*)d_in[20];
    const float* b2 = (const float*)d_in[21];
    const float* w3 = (const float*)d_in[22];
    const float* b3 = (const float*)d_in[23];

    const int n_links = in_sizes[0];
    const int n_nodes = in_sizes[1];
    const int n_paths = in_sizes[2];

    // Workspace layout (floats), all 16B-aligned for the given sizes.
    float* ws = (float*)d_ws;
    float* link_state = ws;                                   // 4*n_links
    float* node_state = link_state + (size_t)4 * n_links;     // 4*n_nodes
    float* path_state = node_state + (size_t)4 * n_nodes;     // 2*n_paths
    float* m_links    = path_state + (size_t)2 * n_paths;     // 2*n_links
    float* m_nodes    = m_links    + (size_t)2 * n_links;     // 2*n_nodes

    const int B = 256;
    hipLaunchKernelGGL(comnet_init_kernel, dim3((n_paths + B - 1) / B), dim3(B),
                       0, stream, cap, qs, traffic, link_state, node_state,
                       path_state, m_links, m_nodes, n_links, n_nodes, n_paths);

    for (int t = 0; t < 3; ++t) {
        hipLaunchKernelGGL(comnet_path_rnn_kernel,
                           dim3((n_paths + B - 1) / B), dim3(B), 0, stream,
                           node_state, link_state, path_state, nodes, links,
                           m_nodes, m_links, pk, pr, pb, n_paths);
        hipLaunchKernelGGL(comnet_edge_node_update_kernel,
                           dim3((n_links + n_nodes + B - 1) / B), dim3(B), 0, stream,
                           link_state, node_state, m_links, m_nodes,
                           ek, er, eb, nk, nr, nb, n_links, n_nodes);
    }

    // 16 paths per wave, 8 waves per block -> 128 paths per block
    hipLaunchKernelGGL(comnet_readout_wmma_kernel,
                       dim3((n_paths + 127) / 128), dim3(B), 0, stream,
                       path_state, w1, b1, w2, b2, w3, b3,
                       (float*)d_out, n_paths);
}